// GCN_Mid_32873679684169
// MI455X (gfx1250) — compile-verified
//
#include <hip/hip_runtime.h>

typedef __attribute__((ext_vector_type(2))) float v2f;
typedef __attribute__((ext_vector_type(8))) float v8f;

#define D 128            // feature/embedding dim (D_FEAT = D_EMB = 128)
#define LDS_STRIDE 132   // 16x128 A tile padded: stride%64==4 -> conflict-free frags

// ---------------------------------------------------------------------------
// CDNA5 async global->LDS copy (ASYNCcnt-tracked), 16B granule
// ---------------------------------------------------------------------------
__device__ __forceinline__ void async_copy_b128(unsigned lds_addr, const void* gptr) {
    asm volatile("global_load_async_to_lds_b128 %0, %1, off"
                 :: "v"(lds_addr), "v"((unsigned long long)(size_t)gptr)
                 : "memory");
}
__device__ __forceinline__ void wait_async0() {
    asm volatile("s_wait_asynccnt 0" ::: "memory");
}

// ---------------------------------------------------------------------------
// CSR construction: zero counts -> count -> exclusive scan -> cursor copy -> scatter
// ---------------------------------------------------------------------------
__global__ void zero_int_kernel(int* __restrict__ p, int n) {
    int i = blockIdx.x * blockDim.x + threadIdx.x;
    if (i < n) p[i] = 0;
}

__global__ void count_kernel(const int* __restrict__ row, int* __restrict__ cnt, int e) {
    int i = blockIdx.x * blockDim.x + threadIdx.x;
    if (i < e) atomicAdd(&cnt[row[i]], 1);
}

// Single-block Hillis-Steele scan over chunks of 1024 (n ~ 50k -> ~49 chunks).
__global__ void __launch_bounds__(1024)
scan_kernel(const int* __restrict__ cnt, int* __restrict__ ptr, int n) {
    __shared__ int sm[1024];
    __shared__ int carry_s;
    const int tid = threadIdx.x;
    if (tid == 0) carry_s = 0;
    __syncthreads();
    for (int base = 0; base < n; base += 1024) {
        int idx = base + tid;
        int v = (idx < n) ? cnt[idx] : 0;
        sm[tid] = v;
        __syncthreads();
        for (int off = 1; off < 1024; off <<= 1) {
            int t = (tid >= off) ? sm[tid - off] : 0;
            __syncthreads();
            sm[tid] += t;
            __syncthreads();
        }
        if (idx < n) ptr[idx] = carry_s + sm[tid] - v;   // exclusive
        __syncthreads();
        if (tid == 1023) carry_s += sm[1023];
        __syncthreads();
    }
    if (tid == 0) ptr[n] = carry_s;
}

__global__ void copy_int_kernel(const int* __restrict__ src, int* __restrict__ dst, int n) {
    int i = blockIdx.x * blockDim.x + threadIdx.x;
    if (i < n) dst[i] = src[i];
}

__global__ void scatter_kernel(const int* __restrict__ row, const int* __restrict__ col,
                               const float* __restrict__ val, int* __restrict__ cursor,
                               int* __restrict__ ccol, float* __restrict__ cval, int e) {
    int i = blockIdx.x * blockDim.x + threadIdx.x;
    if (i < e) {
        int p = atomicAdd(&cursor[row[i]], 1);
        ccol[p] = col[i];
        cval[p] = val[i];
    }
}

// ---------------------------------------------------------------------------
// CSR SpMM, gather-only: one wave32 per row, lane owns 4 contiguous cols.
//   y[r,:] = scale * sum_e val[e]*x[col[e],:]  (+ addScale*addend[r,:])
// No atomics, no accumulator zero-init pass.
// ---------------------------------------------------------------------------
__global__ void __launch_bounds__(256)
csr_spmm_kernel(const int* __restrict__ ptr, const int* __restrict__ ccol,
                const float* __restrict__ cval, const float* __restrict__ x,
                const float* __restrict__ addend, float* __restrict__ y,
                int n_rows, float scale, float addScale) {
    int gid    = blockIdx.x * blockDim.x + threadIdx.x;
    int wid    = gid >> 5;
    int lane   = gid & 31;
    int wtotal = (gridDim.x * blockDim.x) >> 5;

    for (int r = wid; r < n_rows; r += wtotal) {
        int e0 = ptr[r], e1 = ptr[r + 1];
        float4 acc = make_float4(0.f, 0.f, 0.f, 0.f);
        for (int e = e0; e < e1; ++e) {
            int   c = ccol[e];
            float v = cval[e];
            float4 xv = reinterpret_cast<const float4*>(x + (size_t)c * D)[lane];
            acc.x += v * xv.x;  acc.y += v * xv.y;
            acc.z += v * xv.z;  acc.w += v * xv.w;
        }
        float4 res = make_float4(scale * acc.x, scale * acc.y,
                                 scale * acc.z, scale * acc.w);
        if (addend) {
            float4 av = reinterpret_cast<const float4*>(addend + (size_t)r * D)[lane];
            res.x += addScale * av.x;  res.y += addScale * av.y;
            res.z += addScale * av.z;  res.w += addScale * av.w;
        }
        reinterpret_cast<float4*>(y + (size_t)r * D)[lane] = res;
    }
}

// ---------------------------------------------------------------------------
// out[M,128] = A[M,128] @ W[128,128] via V_WMMA_F32_16X16X4_F32.
// Block = 8 waves; each block owns an M tile (grid-strided): A tile staged once
// into LDS via async copy, waves split the 8 N-tiles with B register-resident.
//
// Layouts per CDNA5 ISA 7.12.2 (wave32):
//   A (16x4, f32):  lane l -> row l%16;  VGPR0/1 = K = 2*(l/16) + {0,1}
//   B (4x16, f32):  lane l -> col l%16;  VGPR0/1 = K = 2*(l/16) + {0,1}
//   C/D (16x16):    lane l -> col l%16;  VGPR j  = row j + 8*(l/16)
// ---------------------------------------------------------------------------
__global__ void __launch_bounds__(256)
wmma_gemm_kernel(const float* __restrict__ A, const float* __restrict__ W,
                 float* __restrict__ out, int m_tiles) {
    __shared__ float atile[16 * LDS_STRIDE];
    const int tid   = threadIdx.x;
    const int lane  = tid & 31;
    const int half  = lane >> 4;        // 0: K pair {0,1}; 1: K pair {2,3}
    const int l16   = lane & 15;
    const int nBase = (tid >> 5) << 4;  // wave id * 16 -> N tile

    // Preload all B fragments for this wave's N-tile (K = 0..127 step 4).
    v2f b[32];
#pragma unroll
    for (int kk = 0; kk < 32; ++kk) {
        int k0 = kk * 4 + half * 2;
        b[kk].x = W[(size_t)(k0 + 0) * D + nBase + l16];
        b[kk].y = W[(size_t)(k0 + 1) * D + nBase + l16];
    }

    for (int mt = blockIdx.x; mt < m_tiles; mt += gridDim.x) {
        // Stage 16x128 A tile: 512 x 16B chunks; thread t copies chunks t, t+256.
        {
            int r0 = tid >> 5,          c0 = (tid & 31) * 4;
            int r1 = (tid + 256) >> 5,  c1 = (tid & 31) * 4;
            async_copy_b128((unsigned)(size_t)&atile[r0 * LDS_STRIDE + c0],
                            A + (size_t)(mt * 16 + r0) * D + c0);
            async_copy_b128((unsigned)(size_t)&atile[r1 * LDS_STRIDE + c1],
                            A + (size_t)(mt * 16 + r1) * D + c1);
            wait_async0();
        }
        __syncthreads();

        v8f acc = {};
#pragma unroll
        for (int kk = 0; kk < 32; ++kk) {
            int k0 = kk * 4 + half * 2;
            const float* ap = &atile[l16 * LDS_STRIDE + k0];
            v2f a;
            a.x = ap[0];
            a.y = ap[1];
            acc = __builtin_amdgcn_wmma_f32_16x16x4_f32(
                false, a, false, b[kk], (short)0, acc, false, false);
        }

        float* orow = out + (size_t)(mt * 16 + half * 8) * D + nBase + l16;
#pragma unroll
        for (int j = 0; j < 8; ++j)
            orow[(size_t)j * D] = acc[j];

        __syncthreads();   // LDS reuse fence before next tile's async writes
    }
}

// ---------------------------------------------------------------------------
// inputs: [0] feature [N,128] f32, [1] adj_row [E] i32, [2] adj_col [E] i32,
//         [3] adj_val [E] f32,     [4] weight [128,128] f32
// output: [N,128] f32
// ---------------------------------------------------------------------------
extern "C" void kernel_launch(void* const* d_in, const int* in_sizes, int n_in,
                              void* d_out, int out_size, void* d_ws, size_t ws_size,
                              hipStream_t stream) {
    const float* feature = (const float*)d_in[0];
    const int*   adj_row = (const int*)d_in[1];
    const int*   adj_col = (const int*)d_in[2];
    const float* adj_val = (const float*)d_in[3];
    const float* weight  = (const float*)d_in[4];
    float*       out     = (float*)d_out;

    const int n_nodes = in_sizes[0] / D;
    const int n_edges = in_sizes[1];
    const size_t fe = (size_t)n_nodes * D;

    // Workspace layout (all 4B types; float4 regions are 16B aligned).
    float* h1      = (float*)d_ws;              // adj @ feature
    float* cbuf    = h1 + fe;                   // 0.5*adj@h1 - 0.5*feature
    float* csr_val = cbuf + fe;                 // E floats
    int*   row_ptr = (int*)(csr_val + n_edges); // n+1 ints
    int*   cursor  = row_ptr + (n_nodes + 1);   // n ints (counts, then cursors)
    int*   csr_col = cursor + n_nodes;          // E ints

    const int TB = 256;
    const int nBlk = (n_nodes + TB - 1) / TB;
    const int eBlk = (n_edges + TB - 1) / TB;

    // Build CSR once; reused by both SpMM passes.
    zero_int_kernel<<<nBlk, TB, 0, stream>>>(cursor, n_nodes);
    count_kernel<<<eBlk, TB, 0, stream>>>(adj_row, cursor, n_edges);
    scan_kernel<<<1, 1024, 0, stream>>>(cursor, row_ptr, n_nodes);
    copy_int_kernel<<<nBlk, TB, 0, stream>>>(row_ptr, cursor, n_nodes);
    scatter_kernel<<<eBlk, TB, 0, stream>>>(adj_row, adj_col, adj_val,
                                            cursor, csr_col, csr_val, n_edges);

    // Gather-only SpMM passes (no fp atomics).
    csr_spmm_kernel<<<2048, TB, 0, stream>>>(row_ptr, csr_col, csr_val,
                                             feature, nullptr, h1,
                                             n_nodes, 1.0f, 0.0f);
    csr_spmm_kernel<<<2048, TB, 0, stream>>>(row_ptr, csr_col, csr_val,
                                             h1, feature, cbuf,
                                             n_nodes, 0.5f, -0.5f);

    // Dense projection: 50000 rows = 3125 exact 16-row tiles.
    wmma_gemm_kernel<<<1024, TB, 0, stream>>>(cbuf, weight, out, n_nodes / 16);
}